// OuterProductMean_30313879176037
// MI455X (gfx1250) — compile-verified
//
#include <hip/hip_runtime.h>

typedef float v2f __attribute__((ext_vector_type(2)));
typedef float v4f __attribute__((ext_vector_type(4)));
typedef float v8f __attribute__((ext_vector_type(8)));

constexpr int Bn = 8;          // batches
constexpr int S  = 8192;       // reduction dim
constexpr int I  = 256;        // out rows
constexpr int J  = 256;        // out cols
constexpr int TILE = 128;      // macro tile edge (M and N)
constexpr int KC   = 16;       // K rows staged in LDS per iteration
constexpr int PR   = KC / 2;   // K-pair rows in LDS (float2-interleaved)
constexpr int RS   = 144;      // float2 stride per pair-row: 2*RS mod 64 == 32 -> conflict-free b64
constexpr int SLAB = PR * RS * 2;   // floats per slab buffer

// LDS layout: element (k, col) of a slab sits at float index ((k/2)*RS + col)*2 + (k&1).
// A WMMA f32 fragment (K=2h, K=2h+1 at one col) is then ONE contiguous ds_load_b64.
// Double-buffered: one __syncthreads per stage; stores for slab st overlap WMMAs of slab st-1.

__global__ __launch_bounds__(256)
void opm_gemm(const float* __restrict__ A, const float* __restrict__ Bm,
              float* __restrict__ out, int SC, float scale)
{
    __shared__ __align__(16) float As[2][SLAB];
    __shared__ __align__(16) float Bs[2][SLAB];

    const int tid  = threadIdx.x;
    const int lane = tid & 31;
    const int wave = tid >> 5;
    const int lm   = lane & 15;
    const int half = lane >> 4;

    const int ti = blockIdx.x >> 1;     // 0..1 : M tile
    const int tj = blockIdx.x & 1;      // 0..1 : N tile
    const int bb = blockIdx.y;          // batch
    const int ck = blockIdx.z;          // S chunk

    const int i0 = ti * TILE;
    const int j0 = tj * TILE;
    const int s0 = ck * SC;

    const float* aB = A  + (size_t)bb * S * I;
    const float* bB = Bm + (size_t)bb * S * J;

    const int wm = wave >> 1;           // 0..3 along M
    const int wn = wave & 1;            // 0..1 along N
    const int mBase = wm * 32;
    const int nBase = wn * 64;

    v8f acc[2][4];
#pragma unroll
    for (int tm = 0; tm < 2; ++tm)
#pragma unroll
        for (int tn = 0; tn < 4; ++tn)
            acc[tm][tn] = {};

    // Staging: thread q=tid>>5 owns global rows (2q, 2q+1), cols c4..c4+3 of each slab.
    const int q  = tid >> 5;            // pair-row 0..7
    const int c4 = (tid & 31) * 4;      // column offset (floats)

    v4f pa0, pa1, pb0, pb1;
    const int nstages = SC / KC;

    {   // prologue: fetch slab 0 into registers
        const float* aR = aB + (size_t)(s0 + 2 * q) * I + i0 + c4;
        const float* bR = bB + (size_t)(s0 + 2 * q) * J + j0 + c4;
        pa0 = *(const v4f*)aR;  pa1 = *(const v4f*)(aR + I);
        pb0 = *(const v4f*)bR;  pb1 = *(const v4f*)(bR + J);
    }

    for (int st = 0; st < nstages; ++st) {
        const int buf = st & 1;
        {   // store slab st: interleave rows (2q,2q+1) into float2 pairs, 2x b128 per slab
            float* da = &As[buf][(q * RS + c4) * 2];
            float* db = &Bs[buf][(q * RS + c4) * 2];
            v4f alo = {pa0.x, pa1.x, pa0.y, pa1.y};
            v4f ahi = {pa0.z, pa1.z, pa0.w, pa1.w};
            v4f blo = {pb0.x, pb1.x, pb0.y, pb1.y};
            v4f bhi = {pb0.z, pb1.z, pb0.w, pb1.w};
            *(v4f*)da       = alo;  *(v4f*)(da + 4) = ahi;
            *(v4f*)db       = blo;  *(v4f*)(db + 4) = bhi;
        }

        if (st + 1 < nstages) {  // issue next slab's global loads; latency spans the barrier + compute
            const int sN = s0 + (st + 1) * KC;
            const float* aR = aB + (size_t)(sN + 2 * q) * I + i0 + c4;
            const float* bR = bB + (size_t)(sN + 2 * q) * J + j0 + c4;
            pa0 = *(const v4f*)aR;  pa1 = *(const v4f*)(aR + I);
            pb0 = *(const v4f*)bR;  pb1 = *(const v4f*)(bR + J);
        }
        if (st + 2 < nstages) {  // L2 prefetch two stages ahead (global_prefetch_b8)
            const int sP = s0 + (st + 2) * KC;
            __builtin_prefetch(aB + (size_t)(sP + 2 * q) * I + i0 + c4, 0, 1);
            __builtin_prefetch(bB + (size_t)(sP + 2 * q) * J + j0 + c4, 0, 1);
        }

        __syncthreads();         // slab st visible to all waves; also fences buffer reuse

#pragma unroll
        for (int kk = 0; kk < KC; kk += 4) {
            // pair index for this lane: lanes 0-15 -> kk/2, lanes 16-31 -> kk/2+1
            const int p0 = (kk >> 1) + half;
            v2f aF[2], bF[4];
#pragma unroll
            for (int tm = 0; tm < 2; ++tm)
                aF[tm] = *(const v2f*)&As[buf][(p0 * RS + mBase + tm * 16 + lm) * 2];
#pragma unroll
            for (int tn = 0; tn < 4; ++tn)
                bF[tn] = *(const v2f*)&Bs[buf][(p0 * RS + nBase + tn * 16 + lm) * 2];
#pragma unroll
            for (int tm = 0; tm < 2; ++tm)
#pragma unroll
                for (int tn = 0; tn < 4; ++tn)
                    acc[tm][tn] = __builtin_amdgcn_wmma_f32_16x16x4_f32(
                        false, aF[tm], false, bF[tn],
                        (short)0, acc[tm][tn], false, false);
        }
    }

    // C/D layout: VGPR v, lanes 0-15 -> M=v, lanes 16-31 -> M=v+8
    float* oT = out + ((size_t)ck * Bn + bb) * I * J;
#pragma unroll
    for (int tm = 0; tm < 2; ++tm) {
#pragma unroll
        for (int tn = 0; tn < 4; ++tn) {
            const int gj = j0 + nBase + tn * 16 + lm;
#pragma unroll
            for (int v = 0; v < 8; ++v) {
                const int gi = i0 + mBase + tm * 16 + v + 8 * half;
                oT[(size_t)gi * J + gj] = acc[tm][tn][v] * scale;
            }
        }
    }
}

// Deterministic chunk reduction: out = (1/S) * sum_c part[c]
__global__ __launch_bounds__(256)
void opm_reduce(const float* __restrict__ part, float* __restrict__ out, int NS)
{
    const size_t n = (size_t)Bn * I * J;
    size_t idx = (size_t)blockIdx.x * blockDim.x + threadIdx.x;
    if (idx >= n) return;
    float s = 0.f;
    for (int c = 0; c < NS; ++c) s += part[(size_t)c * n + idx];
    out[idx] = s * (1.0f / (float)S);
}

extern "C" void kernel_launch(void* const* d_in, const int* in_sizes, int n_in,
                              void* d_out, int out_size, void* d_ws, size_t ws_size,
                              hipStream_t stream)
{
    const float* a = (const float*)d_in[0];
    const float* b = (const float*)d_in[1];
    float* out = (float*)d_out;

    // Pick S-split from available workspace (powers of two divide S=8192, SC%KC==0).
    const size_t perChunk = (size_t)Bn * I * J * sizeof(float);
    int NS = 16;
    while (NS > 1 && (size_t)NS * perChunk > ws_size) NS >>= 1;

    dim3 block(256);
    if (NS == 1) {
        dim3 grid(4, Bn, 1);
        opm_gemm<<<grid, block, 0, stream>>>(a, b, out, S, 1.0f / (float)S);
    } else {
        float* part = (float*)d_ws;
        dim3 grid(4, Bn, NS);
        opm_gemm<<<grid, block, 0, stream>>>(a, b, part, S / NS, 1.0f);
        const size_t n = (size_t)Bn * I * J;
        opm_reduce<<<(unsigned)((n + 255) / 256), 256, 0, stream>>>(part, out, NS);
    }
}